// BinarizedArithmeticModule_6468220748632
// MI455X (gfx1250) — compile-verified
//
#include <hip/hip_runtime.h>
#include <cstdint>

// ---------------------------------------------------------------------------
// out = unbinarize( (tanh(W_hat) * sigmoid(M_hat)) @ binarize(inputs) )
//   inputs: 1024 f32  -> 32768 bits (K)
//   W_hat/M_hat: [4096, 32768] f32   (the ~1.07 GB stream: HBM-bound, ~46us)
//   out: 128 f32 (bit-packed)
// ---------------------------------------------------------------------------

typedef __attribute__((ext_vector_type(2))) float v2f;
typedef __attribute__((ext_vector_type(4))) float v4f;
typedef __attribute__((ext_vector_type(8))) float v8f;

#define K_TOTAL        32768
#define M_TOTAL        4096
#define ROWS_PER_BLOCK 16
#define KCHUNKS        8
#define KC_COLS        (K_TOTAL / KCHUNKS)     // 4096 columns per block
#define TILE_COLS      512
#define TILES_PER_KC   (KC_COLS / TILE_COLS)   // 8
#define NWAVES         8
#define WAVE_COLS      (TILE_COLS / NWAVES)    // 64
#define LDS_STRIDE     (TILE_COLS + 4)         // pad -> conflict-free A-frag reads

__device__ __forceinline__ float nac_w(float wh, float mh) {
    // tanh(wh) * sigmoid(mh) using hw v_exp_f32 / v_rcp_f32
    float t = 1.0f - 2.0f * __builtin_amdgcn_rcpf(__expf(2.0f * wh) + 1.0f);
    float s = __builtin_amdgcn_rcpf(1.0f + __expf(-mh));
    return t * s;
}

extern "C" __global__ void __launch_bounds__(256)
nac_gemv_wmma_kernel(const uint32_t* __restrict__ inp_u32,
                     const float* __restrict__ W_hat,
                     const float* __restrict__ M_hat,
                     float* __restrict__ part)
{
    __shared__ float w_lds[ROWS_PER_BLOCK * LDS_STRIDE];  // 16x(512+4) f32
    __shared__ float bits_lds[KC_COLS];                   // this block's 4096 bit values
    __shared__ float red[NWAVES * ROWS_PER_BLOCK];

    const int tid     = threadIdx.x;
    const int lane    = tid & 31;
    const int wave    = tid >> 5;
    const int rowBase = blockIdx.x * ROWS_PER_BLOCK;   // 0..4080
    const int kc      = blockIdx.y;                    // 0..7
    const int colBase = kc * KC_COLS;

    // Stage binarized input bits for this block's column range.
    // numpy unpackbits(view(uint8)) order per 32-bit word: shift = (t & 31) ^ 7
    for (int i = tid; i < KC_COLS; i += 256) {
        int c = colBase + i;
        uint32_t u = inp_u32[c >> 5];
        bits_lds[i] = (float)((u >> ((c & 31) ^ 7)) & 1u);
    }

    v8f acc = {};  // 16x16 f32 C/D accumulator; only column N=0 is meaningful

    // A-fragment (16x4 f32) addressing: lane<16 -> row=lane,   K=0,1
    //                                   lane>=16 -> row=lane-16, K=2,3
    const int  aRow   = lane & 15;
    const int  aKoff  = (lane >> 4) << 1;              // 0 or 2
    const float bMask = (lane == 0 || lane == 16) ? 1.0f : 0.0f;  // N=0 only

    __syncthreads();

    for (int tile = 0; tile < TILES_PER_KC; ++tile) {
        // ---- global stream (coalesced b128) -> elementwise NAC weight -> LDS
        const int tileCol0 = colBase + tile * TILE_COLS;
        #pragma unroll
        for (int i = 0; i < (ROWS_PER_BLOCK * TILE_COLS / 4) / 256; ++i) {  // 8 iters
            int flat4 = tid + i * 256;
            int row   = flat4 >> 7;        // 128 float4 per row
            int col4  = flat4 & 127;
            size_t g  = (size_t)(rowBase + row) * K_TOTAL + (size_t)tileCol0 + col4 * 4;
            v4f wh = *(const v4f*)(W_hat + g);
            v4f mh = *(const v4f*)(M_hat + g);
            v4f w;
            w.x = nac_w(wh.x, mh.x);
            w.y = nac_w(wh.y, mh.y);
            w.z = nac_w(wh.z, mh.z);
            w.w = nac_w(wh.w, mh.w);
            *(v4f*)(w_lds + row * LDS_STRIDE + col4 * 4) = w;
        }
        __syncthreads();

        // ---- each wave: 16 x WMMA f32 16x16x4 over its 64-column slice
        #pragma unroll
        for (int kk = 0; kk < WAVE_COLS / 4; ++kk) {
            int cloc = wave * WAVE_COLS + kk * 4;      // column within tile
            v2f a = *(const v2f*)(w_lds + aRow * LDS_STRIDE + cloc + aKoff);
            // B (4x16): lane0 holds K=0,1 / lane16 holds K=2,3 for N=0; rest zero
            v2f braw = *(const v2f*)(bits_lds + tile * TILE_COLS + cloc + aKoff);
            v2f b;
            b.x = braw.x * bMask;
            b.y = braw.y * bMask;
            acc = __builtin_amdgcn_wmma_f32_16x16x4_f32(
                      /*neg_a=*/false, a, /*neg_b=*/false, b,
                      /*c_mod=*/(short)0, acc, /*reuse_a=*/false, /*reuse_b=*/false);
        }
        __syncthreads();   // protect w_lds before next tile's stores
    }

    // ---- extract column N=0 of C: lane0 v[g]=row g, lane16 v[g]=row 8+g
    if (lane == 0) {
        #pragma unroll
        for (int g = 0; g < 8; ++g) red[wave * 16 + g] = acc[g];
    }
    if (lane == 16) {
        #pragma unroll
        for (int g = 0; g < 8; ++g) red[wave * 16 + 8 + g] = acc[g];
    }
    __syncthreads();

    if (tid < ROWS_PER_BLOCK) {
        float s = 0.0f;
        #pragma unroll
        for (int w = 0; w < NWAVES; ++w) s += red[w * 16 + tid];
        part[(size_t)(rowBase + tid) * KCHUNKS + kc] = s;   // deterministic, no atomics
    }
}

// Sum K-chunk partials per row, round(RNE)+clamp to a bit, pack 32 bits/word.
extern "C" __global__ void __launch_bounds__(128)
nac_pack_kernel(const float* __restrict__ part, float* __restrict__ out)
{
    int o = threadIdx.x;            // 0..127 output words
    if (o >= 128) return;
    uint32_t word = 0;
    for (int t = 0; t < 32; ++t) {
        int r = o * 32 + t;
        float s = 0.0f;
        #pragma unroll
        for (int j = 0; j < KCHUNKS; ++j) s += part[(size_t)r * KCHUNKS + j];
        float b = fminf(fmaxf(rintf(s), 0.0f), 1.0f);   // jnp.round == RNE, then clip
        word |= ((uint32_t)b) << (t ^ 7);
    }
    out[o] = __uint_as_float(word);
}

extern "C" void kernel_launch(void* const* d_in, const int* in_sizes, int n_in,
                              void* d_out, int out_size, void* d_ws, size_t ws_size,
                              hipStream_t stream)
{
    const uint32_t* inp = (const uint32_t*)d_in[0];   // inputs, 1024 f32 viewed as bits
    const float*    Wh  = (const float*)d_in[1];      // W_hat [4096, 32768]
    const float*    Mh  = (const float*)d_in[2];      // M_hat [4096, 32768]
    float* part = (float*)d_ws;                       // 4096*8 f32 partials (128 KB)
    float* out  = (float*)d_out;                      // 128 f32

    dim3 grid(M_TOTAL / ROWS_PER_BLOCK, KCHUNKS);     // 256 x 8 blocks
    nac_gemv_wmma_kernel<<<grid, 256, 0, stream>>>(inp, Wh, Mh, part);
    nac_pack_kernel<<<1, 128, 0, stream>>>(part, out);
}